// AttentionAggregator_18923625906526
// MI455X (gfx1250) — compile-verified
//
#include <hip/hip_runtime.h>

typedef __attribute__((ext_vector_type(2))) float v2f;
typedef __attribute__((ext_vector_type(8))) float v8f;

#define D_IN   256
#define D_OUT  128
#define S_NB   32
#define ALPHA  0.2f
#define KCHUNK 64          // K rows of W staged in LDS per chunk (32 KB)

__device__ __forceinline__ float eluf(float v) {
    return v > 0.f ? v : (__expf(v) - 1.f);
}

// ---------------------------------------------------------------------------
// Kernel 1: xw = x @ W via V_WMMA_F32_16X16X4_F32, with W staged in LDS as
// interleaved K-pairs so each B fragment is one ds_load_b64. Also fuses the
// per-row attention dot products (eself = xw.a_self, fneib = xw.a_neib) out
// of the accumulators, eliminating a separate 25.6MB pass.
// One wave -> 16 rows x 128 cols (8 C tiles). Block = 8 waves = 128 rows.
// ---------------------------------------------------------------------------
__global__ __launch_bounds__(256) void gemm_xw_kernel(
    const float* __restrict__ x, const float* __restrict__ W,
    const float* __restrict__ avec,
    float* __restrict__ xw, float* __restrict__ eself,
    float* __restrict__ fneib, int N) {
    // lw packed as v2f: entry [p*128 + c] = {W[kc+2p][c], W[kc+2p+1][c]}
    __shared__ float lw[KCHUNK * D_OUT];     // 32 KB

    const int wave = threadIdx.x >> 5;
    const int lane = threadIdx.x & 31;
    const int half = lane >> 4;              // 0: K={0,1}, 1: K={2,3}
    const int lr   = lane & 15;              // row (A) / col (B,C) within tile
    const int rowBase = (blockIdx.x * 8 + wave) * 16;

    int arow = rowBase + lr;
    if (arow > N - 1) arow = N - 1;          // clamp loads: EXEC stays all-1s
    const float* xr = x + (size_t)arow * D_IN;
    const int kh = 2 * half;

    v8f c[8];
#pragma unroll
    for (int t = 0; t < 8; ++t)
        c[t] = (v8f){0.f, 0.f, 0.f, 0.f, 0.f, 0.f, 0.f, 0.f};

    for (int kc = 0; kc < D_IN; kc += KCHUNK) {
        __syncthreads();
        // cooperative stage: 64 K-rows -> 32 pairs x 128 cols of v2f
        for (int l = threadIdx.x; l < (KCHUNK / 2) * D_OUT; l += 256) {
            const int p  = l >> 7;           // pair index 0..31
            const int cI = l & 127;          // column
            v2f w2;
            w2.x = W[(size_t)(kc + 2 * p)     * D_OUT + cI];
            w2.y = W[(size_t)(kc + 2 * p + 1) * D_OUT + cI];
            ((v2f*)lw)[l] = w2;
        }
        __syncthreads();

        for (int k = kc; k < kc + KCHUNK; k += 4) {
            v2f a;
            a.x = xr[k + kh];
            a.y = xr[k + kh + 1];
            const int prow = (k - kc + kh) >> 1;   // local pair row
#pragma unroll
            for (int t = 0; t < 8; ++t) {
                v2f b = ((const v2f*)lw)[prow * 128 + t * 16 + lr];
                c[t] = __builtin_amdgcn_wmma_f32_16x16x4_f32(
                    false, a, false, b, (short)0, c[t], false, false);
            }
        }
    }

    // ---- store xw tile: C VGPRr -> M = r + 8*half, N-col = t*16 + lr ----
#pragma unroll
    for (int t = 0; t < 8; ++t) {
#pragma unroll
        for (int r = 0; r < 8; ++r) {
            const int row = rowBase + half * 8 + r;
            if (row < N)
                xw[(size_t)row * D_OUT + t * 16 + lr] = c[t][r];
        }
    }

    // ---- fused dot products straight from the accumulators ----
    float as[8], an[8];
#pragma unroll
    for (int t = 0; t < 8; ++t) {
        as[t] = avec[t * 16 + lr];           // a_self
        an[t] = avec[D_OUT + t * 16 + lr];   // a_neib
    }
    float s[8], f[8];
#pragma unroll
    for (int r = 0; r < 8; ++r) {
        float ss = 0.f, ff = 0.f;
#pragma unroll
        for (int t = 0; t < 8; ++t) {
            ss += c[t][r] * as[t];
            ff += c[t][r] * an[t];
        }
        // reduce over the 16 lanes of this half (xor masks < 16 stay in-half)
#pragma unroll
        for (int off = 8; off > 0; off >>= 1) {
            ss += __shfl_xor(ss, off, 32);
            ff += __shfl_xor(ff, off, 32);
        }
        s[r] = ss; f[r] = ff;
    }
    if (lr == 0) {
#pragma unroll
        for (int r = 0; r < 8; ++r) {
            const int row = rowBase + half * 8 + r;
            if (row < N) { eself[row] = s[r]; fneib[row] = f[r]; }
        }
    }
}

// ---------------------------------------------------------------------------
// Kernel 2: fused gather + LeakyReLU + softmax + weighted aggregate + ELU.
// One wave per node. Lane s handles neighbor s for the logit/softmax; for the
// aggregation each lane owns a float4 of the 128-dim row and sweeps all 32
// neighbor rows with broadcast weights (coalesced 512B row reads, L2-hot
// since xw is 25.6MB << 192MB L2).
// ---------------------------------------------------------------------------
__global__ __launch_bounds__(256) void attn_kernel(
    const float* __restrict__ xw, const int* __restrict__ neibs,
    const float* __restrict__ eself, const float* __restrict__ fneib,
    float* __restrict__ out, int N) {
    const int w    = (int)((blockIdx.x * blockDim.x + threadIdx.x) >> 5);
    const int lane = threadIdx.x & 31;
    if (w >= N) return;

    const int idx = neibs[(size_t)w * S_NB + lane];
    float e = eself[w] + fneib[idx];
    e = e > 0.f ? e : ALPHA * e;                  // LeakyReLU(0.2)

    float m = e;                                  // softmax over 32 lanes
#pragma unroll
    for (int off = 16; off > 0; off >>= 1)
        m = fmaxf(m, __shfl_xor(m, off, 32));
    float p = __expf(e - m);
    float sum = p;
#pragma unroll
    for (int off = 16; off > 0; off >>= 1)
        sum += __shfl_xor(sum, off, 32);
    const float att = p / sum;

    float4 acc = {0.f, 0.f, 0.f, 0.f};
#pragma unroll 4
    for (int s = 0; s < S_NB; ++s) {
        const float a_s  = __shfl(att, s, 32);
        const int   id_s = __shfl(idx, s, 32);
        float4 v = ((const float4*)(xw + (size_t)id_s * D_OUT))[lane];
        acc.x += a_s * v.x;
        acc.y += a_s * v.y;
        acc.z += a_s * v.z;
        acc.w += a_s * v.w;
    }

    float4 own = ((const float4*)(xw + (size_t)w * D_OUT))[lane];
    float* orow = out + (size_t)w * (2 * D_OUT);
    float4 o1, o2;
    o1.x = eluf(own.x); o1.y = eluf(own.y); o1.z = eluf(own.z); o1.w = eluf(own.w);
    o2.x = eluf(acc.x); o2.y = eluf(acc.y); o2.z = eluf(acc.z); o2.w = eluf(acc.w);
    ((float4*)orow)[lane]           = o1;   // out[:, 0:128]   = elu(xw)
    ((float4*)(orow + D_OUT))[lane] = o2;   // out[:, 128:256] = elu(h_prime)
}

// ---------------------------------------------------------------------------
// inputs (setup_inputs order): x (N,256) f32 | neibs (N,32) int | W (256,128)
// f32 | a (256,1) f32.  out: (N,256) f32.  ws: xw (N*128) + eself (N) + fneib (N).
// ---------------------------------------------------------------------------
extern "C" void kernel_launch(void* const* d_in, const int* in_sizes, int n_in,
                              void* d_out, int out_size, void* d_ws, size_t ws_size,
                              hipStream_t stream) {
    const float* x     = (const float*)d_in[0];
    const int*   neibs = (const int*)d_in[1];
    const float* W     = (const float*)d_in[2];
    const float* a     = (const float*)d_in[3];
    float* out = (float*)d_out;

    const int N = in_sizes[0] / D_IN;

    float* xw    = (float*)d_ws;
    float* eself = xw + (size_t)N * D_OUT;
    float* fneib = eself + N;

    gemm_xw_kernel<<<(N + 127) / 128, 256, 0, stream>>>(x, W, a, xw, eself, fneib, N);
    attn_kernel<<<(N + 7) / 8, 256, 0, stream>>>(xw, neibs, eself, fneib, out, N);
}